// Attention_45835890982922
// MI455X (gfx1250) — compile-verified
//
#include <hip/hip_runtime.h>
#include <hip/hip_bf16.h>
#include <stdint.h>

// ---------------------------------------------------------------- constants
constexpr int NB   = 4;      // batch
constexpr int NSEQ = 2048;   // sequence length
constexpr int NC   = 768;    // channels
constexpr int NH   = 12;     // heads
constexpr int ND   = 64;     // head dim
constexpr int C3   = 3 * NC; // 2304
constexpr int KC   = 32;     // GEMM k-chunk
constexpr int PA   = 36;     // LDS pitch for A tiles (16B aligned, conflict-free frag reads)
constexpr int PB   = 68;     // LDS pitch for B/K/V/P tiles (16B aligned)
constexpr float LN_EPS   = 1e-5f;
constexpr float QK_SCALE = 0.125f;   // 64^-0.5

typedef float v2f __attribute__((ext_vector_type(2)));
typedef float v8f __attribute__((ext_vector_type(8)));

// ------------------------------------------------- CDNA5 async copy to LDS
#define USE_ASYNC_LDS 1

__device__ __forceinline__ void cp_to_lds16(float* lds_dst, const float* gsrc) {
#if USE_ASYNC_LDS
  // low 32 bits of a flat LDS address == workgroup-relative LDS byte offset
  uint32_t loff = (uint32_t)(uintptr_t)lds_dst;
  asm volatile("global_load_async_to_lds_b128 %0, %1, off"
               :: "v"(loff), "v"(gsrc) : "memory");
#else
  *(float4*)lds_dst = *(const float4*)gsrc;
#endif
}

// Wait until at most N async ops remain outstanding (ASYNCcnt decrements in
// order, so waiting <= N_next guarantees the previous chunk has landed while
// the next chunk's loads stay in flight).
template <int N>
__device__ __forceinline__ void wait_async_le() {
#if USE_ASYNC_LDS
  asm volatile("s_wait_asynccnt %0" :: "i"(N) : "memory");
#endif
}

// ------------------------------------------------------------- WMMA helper
__device__ __forceinline__ v8f wmma4(v2f a, v2f b, v8f c) {
  return __builtin_amdgcn_wmma_f32_16x16x4_f32(false, a, false, b,
                                               (short)0, c, false, false);
}

// ----------------------------------------------------- 16-lane half reduces
__device__ __forceinline__ float hsum16(float v) {
  v += __shfl_xor(v, 1, 32);
  v += __shfl_xor(v, 2, 32);
  v += __shfl_xor(v, 4, 32);
  v += __shfl_xor(v, 8, 32);
  return v;
}
__device__ __forceinline__ float hmax16(float v) {
  v = fmaxf(v, __shfl_xor(v, 1, 32));
  v = fmaxf(v, __shfl_xor(v, 2, 32));
  v = fmaxf(v, __shfl_xor(v, 4, 32));
  v = fmaxf(v, __shfl_xor(v, 8, 32));
  return v;
}

// -------------------------------------------- shared 128x64 GEMM main loop
// Block = 256 threads (8 waves); wave w owns rows [w*16, w*16+16) of the tile.
// Double-buffered LDS, async global->LDS overlapped with WMMA compute.
constexpr int ABUF = 128 * PA;   // floats per A buffer
constexpr int BBUF = KC * PB;    // floats per B buffer

__device__ __forceinline__ void gemm_issue_loads(const float* __restrict__ Ag, int lda,
                                                 const float* __restrict__ Bg, int ldb,
                                                 int k0, float* lA, float* lB) {
  const int tid = threadIdx.x;
#pragma unroll
  for (int i = 0; i < 4; ++i) {               // A: 128 x KC = 1024 vec4
    int p = tid + i * 256;
    int row = p >> 3, cv = (p & 7) * 4;
    cp_to_lds16(&lA[row * PA + cv], &Ag[(size_t)row * lda + k0 + cv]);
  }
#pragma unroll
  for (int i = 0; i < 2; ++i) {               // B: KC x 64 = 512 vec4
    int p = tid + i * 256;
    int row = p >> 4, cv = (p & 15) * 4;
    cp_to_lds16(&lB[row * PB + cv], &Bg[(size_t)(k0 + row) * ldb + cv]);
  }
}

__device__ __forceinline__ void gemm_compute_chunk(const float* lA, const float* lB,
                                                   v8f acc[4], int ar, int half, int ln) {
#pragma unroll
  for (int k = 0; k < KC; k += 4) {
    v2f a;
    a.x = lA[ar * PA + k + half * 2];
    a.y = lA[ar * PA + k + half * 2 + 1];
#pragma unroll
    for (int cb = 0; cb < 4; ++cb) {
      const int col = cb * 16 + ln;
      v2f b;
      b.x = lB[(k + half * 2) * PB + col];
      b.y = lB[(k + half * 2 + 1) * PB + col];
      acc[cb] = wmma4(a, b, acc[cb]);
    }
  }
}

// lA: 2*ABUF floats, lB: 2*BBUF floats (double buffers, selected by offset)
__device__ __forceinline__ void gemm128x64(const float* __restrict__ Ag, int lda,
                                           const float* __restrict__ Bg, int ldb,
                                           int Kdim, float* lA, float* lB,
                                           v8f acc[4]) {
  const int lane = threadIdx.x & 31;
  const int wave = threadIdx.x >> 5;
  const int ln   = lane & 15;
  const int half = lane >> 4;
  const int ar   = wave * 16 + ln;

  const int nch = Kdim / KC;
  gemm_issue_loads(Ag, lda, Bg, ldb, 0, lA, lB);
  for (int c = 0; c < nch; ++c) {
    const int cur = c & 1, nxt = cur ^ 1;
    if (c + 1 < nch) {
      gemm_issue_loads(Ag, lda, Bg, ldb, (c + 1) * KC, lA + nxt * ABUF, lB + nxt * BBUF);
      wait_async_le<6>();   // 6 ops/thread in flight for next chunk; current done
    } else {
      wait_async_le<0>();
    }
    __syncthreads();
    gemm_compute_chunk(lA + cur * ABUF, lB + cur * BBUF, acc, ar, half, ln);
    __syncthreads();        // all readers done -> buffer reusable next iter
  }
}

// ============================================================== kernel 1
// qkv = x @ w_qkv, fused per-head LayerNorm on q/k (+ q scale), relayout to
// Q/K/V in [B,H,N,D]. Grid: (3*H strips of 64 cols, BN/128 row tiles).
__global__ __launch_bounds__(256) void qkv_ln_kernel(
    const float* __restrict__ x, const float* __restrict__ w_qkv,
    const float* __restrict__ qg, const float* __restrict__ qb,
    const float* __restrict__ kg, const float* __restrict__ kbv,
    float* __restrict__ Qo, float* __restrict__ Ko, float* __restrict__ Vo) {
  __shared__ float lA[2 * ABUF];
  __shared__ float lB[2 * BBUF];

  const int strip = blockIdx.x;          // 0..35
  const int r0    = blockIdx.y * 128;    // global row base (b*N + n)
  const int t     = strip / NH;          // 0=q 1=k 2=v
  const int h     = strip % NH;
  const int lane  = threadIdx.x & 31;
  const int wave  = threadIdx.x >> 5;
  const int ln    = lane & 15;
  const int half  = lane >> 4;

  v8f acc[4] = {};
  gemm128x64(x + (size_t)r0 * NC, NC, w_qkv + strip * 64, C3, NC, lA, lB, acc);

  float gamma[4], beta[4];
  float postscale = 1.0f;
  if (t == 0) {
#pragma unroll
    for (int cb = 0; cb < 4; ++cb) { gamma[cb] = qg[cb * 16 + ln]; beta[cb] = qb[cb * 16 + ln]; }
    postscale = QK_SCALE;
  } else if (t == 1) {
#pragma unroll
    for (int cb = 0; cb < 4; ++cb) { gamma[cb] = kg[cb * 16 + ln]; beta[cb] = kbv[cb * 16 + ln]; }
  }
  float* outbase = (t == 0) ? Qo : (t == 1) ? Ko : Vo;

#pragma unroll
  for (int j = 0; j < 8; ++j) {
    const int grow = r0 + wave * 16 + j + half * 8;  // b*N + n
    const int b = grow >> 11;                        // N = 2048
    const int n = grow & (NSEQ - 1);
    float* orow = outbase + ((size_t)((b * NH + h) * NSEQ + n)) * ND;
    float v0 = acc[0][j], v1 = acc[1][j], v2 = acc[2][j], v3 = acc[3][j];
    if (t < 2) {
      float s = hsum16(v0 + v1 + v2 + v3);
      float mu = s * (1.0f / 64.0f);
      float d0 = v0 - mu, d1 = v1 - mu, d2 = v2 - mu, d3 = v3 - mu;
      float q2 = hsum16(d0 * d0 + d1 * d1 + d2 * d2 + d3 * d3);
      float r = rsqrtf(q2 * (1.0f / 64.0f) + LN_EPS);
      orow[0 * 16 + ln] = (d0 * r * gamma[0] + beta[0]) * postscale;
      orow[1 * 16 + ln] = (d1 * r * gamma[1] + beta[1]) * postscale;
      orow[2 * 16 + ln] = (d2 * r * gamma[2] + beta[2]) * postscale;
      orow[3 * 16 + ln] = (d3 * r * gamma[3] + beta[3]) * postscale;
    } else {
      orow[0 * 16 + ln] = v0;
      orow[1 * 16 + ln] = v1;
      orow[2 * 16 + ln] = v2;
      orow[3 * 16 + ln] = v3;
    }
  }
}

// ============================================================== kernel 2
// Flash attention: grid (N/128, B*H); block 256 = 8 waves, 16 q-rows/wave.
// Double-buffered 64-key K/V chunks streamed through LDS via async copies.
constexpr int KVBUF = 64 * PB;   // floats per K (or V) buffer

__global__ __launch_bounds__(256) void attn_kernel(
    const float* __restrict__ Q, const float* __restrict__ K,
    const float* __restrict__ V, float* __restrict__ CTX) {
  __shared__ float lQ[128 * PA];
  __shared__ float lK[2 * KVBUF];
  __shared__ float lV[2 * KVBUF];
  __shared__ float lP[128 * PB];

  const int bh = blockIdx.y;
  const int b  = bh / NH;
  const int h  = bh % NH;
  const int q0 = blockIdx.x * 128;
  const float* Qh = Q + (size_t)bh * NSEQ * ND;
  const float* Kh = K + (size_t)bh * NSEQ * ND;
  const float* Vh = V + (size_t)bh * NSEQ * ND;

  const int tid  = threadIdx.x;
  const int lane = tid & 31;
  const int wave = tid >> 5;
  const int ln   = lane & 15;
  const int half = lane >> 4;
  const int ar   = wave * 16 + ln;

  // Q tile 128x64 (resident for whole kernel): 8 async ops/thread
#pragma unroll
  for (int i = 0; i < 8; ++i) {
    int p = tid + i * 256;
    int row = p >> 4, cv = (p & 15) * 4;
    cp_to_lds16(&lQ[row * PA + cv], &Qh[(size_t)(q0 + row) * ND + cv]);
  }

  auto issue_kv = [&](int kc, int buf) {
    float* dK = lK + buf * KVBUF;
    float* dV = lV + buf * KVBUF;
#pragma unroll
    for (int i = 0; i < 4; ++i) {  // K,V chunks: 64x64 each -> 8 ops/thread
      int p = tid + i * 256;
      int row = p >> 4, cv = (p & 15) * 4;
      cp_to_lds16(&dK[row * PB + cv], &Kh[(size_t)(kc + row) * ND + cv]);
      cp_to_lds16(&dV[row * PB + cv], &Vh[(size_t)(kc + row) * ND + cv]);
    }
  };

  v8f o[4] = {};
  float m_i[8], l_i[8];
#pragma unroll
  for (int j = 0; j < 8; ++j) { m_i[j] = -1e30f; l_i[j] = 0.0f; }

  constexpr int NCHUNK = NSEQ / 64;  // 32
  issue_kv(0, 0);
  for (int c = 0; c < NCHUNK; ++c) {
    const int cur = c & 1, nxt = cur ^ 1;
    if (c + 1 < NCHUNK) {
      issue_kv((c + 1) * 64, nxt);
      wait_async_le<8>();  // next chunk's 8 ops in flight; current (and Q) done
    } else {
      wait_async_le<0>();
    }
    __syncthreads();
    const float* cK = lK + cur * KVBUF;
    const float* cV = lV + cur * KVBUF;

    // S = Q @ K^T over d (q already carries the 1/sqrt(D) scale)
    v8f s[4] = {};
#pragma unroll
    for (int d = 0; d < ND; d += 4) {
      v2f a;
      a.x = lQ[ar * PA + d + half * 2];
      a.y = lQ[ar * PA + d + half * 2 + 1];
#pragma unroll
      for (int cb = 0; cb < 4; ++cb) {
        const int key = cb * 16 + ln;
        v2f bb;
        bb.x = cK[key * PB + d + half * 2];
        bb.y = cK[key * PB + d + half * 2 + 1];
        s[cb] = wmma4(a, bb, s[cb]);
      }
    }

    // online softmax, write P to wave-private LDS rows
#pragma unroll
    for (int j = 0; j < 8; ++j) {
      float x0 = s[0][j], x1 = s[1][j], x2 = s[2][j], x3 = s[3][j];
      float mx = hmax16(fmaxf(fmaxf(x0, x1), fmaxf(x2, x3)));
      float mnew = fmaxf(m_i[j], mx);
      float p0 = __expf(x0 - mnew), p1 = __expf(x1 - mnew);
      float p2 = __expf(x2 - mnew), p3 = __expf(x3 - mnew);
      float rs = hsum16(p0 + p1 + p2 + p3);
      float corr = __expf(m_i[j] - mnew);
      l_i[j] = l_i[j] * corr + rs;
      m_i[j] = mnew;
      o[0][j] *= corr; o[1][j] *= corr; o[2][j] *= corr; o[3][j] *= corr;
      const int prow = wave * 16 + j + half * 8;
      lP[prow * PB + 0 * 16 + ln] = p0;
      lP[prow * PB + 1 * 16 + ln] = p1;
      lP[prow * PB + 2 * 16 + ln] = p2;
      lP[prow * PB + 3 * 16 + ln] = p3;
    }
    // P rows are private to this wave; same-wave LDS ops are in order.

    // O += P @ V over keys
#pragma unroll
    for (int kk = 0; kk < 64; kk += 4) {
      v2f a;
      a.x = lP[ar * PB + kk + half * 2];
      a.y = lP[ar * PB + kk + half * 2 + 1];
#pragma unroll
      for (int cb = 0; cb < 4; ++cb) {
        const int dcol = cb * 16 + ln;
        v2f bb;
        bb.x = cV[(kk + half * 2) * PB + dcol];
        bb.y = cV[(kk + half * 2 + 1) * PB + dcol];
        o[cb] = wmma4(a, bb, o[cb]);
      }
    }
    __syncthreads();  // all waves done with cur K/V -> buffer reusable
  }

  // normalize + write ctx in [B,N,C] layout (head-interleaved) for proj GEMM
#pragma unroll
  for (int j = 0; j < 8; ++j) {
    const int n = q0 + wave * 16 + j + half * 8;
    const float inv = 1.0f / l_i[j];
    float* orow = CTX + ((size_t)(b * NSEQ + n)) * NC + h * ND;
    orow[0 * 16 + ln] = o[0][j] * inv;
    orow[1 * 16 + ln] = o[1][j] * inv;
    orow[2 * 16 + ln] = o[2][j] * inv;
    orow[3 * 16 + ln] = o[3][j] * inv;
  }
}

// ============================================================== kernel 3
// out = ctx @ w_proj + b_proj. Grid (C/64 strips, BN/128 row tiles).
__global__ __launch_bounds__(256) void proj_kernel(
    const float* __restrict__ CTX, const float* __restrict__ W,
    const float* __restrict__ bias, float* __restrict__ out) {
  __shared__ float lA[2 * ABUF];
  __shared__ float lB[2 * BBUF];

  const int strip = blockIdx.x;        // 0..11
  const int r0    = blockIdx.y * 128;
  const int lane  = threadIdx.x & 31;
  const int wave  = threadIdx.x >> 5;
  const int ln    = lane & 15;
  const int half  = lane >> 4;

  v8f acc[4] = {};
  gemm128x64(CTX + (size_t)r0 * NC, NC, W + strip * 64, NC, NC, lA, lB, acc);

  float br[4];
#pragma unroll
  for (int cb = 0; cb < 4; ++cb) br[cb] = bias[strip * 64 + cb * 16 + ln];

#pragma unroll
  for (int j = 0; j < 8; ++j) {
    const int g = r0 + wave * 16 + j + half * 8;
    float* orow = out + (size_t)g * NC + strip * 64;
    orow[0 * 16 + ln] = acc[0][j] + br[0];
    orow[1 * 16 + ln] = acc[1][j] + br[1];
    orow[2 * 16 + ln] = acc[2][j] + br[2];
    orow[3 * 16 + ln] = acc[3][j] + br[3];
  }
}

// ================================================================= launch
extern "C" void kernel_launch(void* const* d_in, const int* in_sizes, int n_in,
                              void* d_out, int out_size, void* d_ws, size_t ws_size,
                              hipStream_t stream) {
  const float* x       = (const float*)d_in[0];
  const float* w_qkv   = (const float*)d_in[1];
  const float* q_gamma = (const float*)d_in[2];
  const float* q_beta  = (const float*)d_in[3];
  const float* k_gamma = (const float*)d_in[4];
  const float* k_beta  = (const float*)d_in[5];
  const float* w_proj  = (const float*)d_in[6];
  const float* b_proj  = (const float*)d_in[7];
  float* out = (float*)d_out;

  const size_t BHND = (size_t)NB * NH * NSEQ * ND;  // 6.29M floats each
  float* Q   = (float*)d_ws;
  float* K   = Q + BHND;
  float* V   = K + BHND;
  float* CTX = V + BHND;

  qkv_ln_kernel<<<dim3(3 * NH, (NB * NSEQ) / 128), 256, 0, stream>>>(
      x, w_qkv, q_gamma, q_beta, k_gamma, k_beta, Q, K, V);
  attn_kernel<<<dim3(NSEQ / 128, NB * NH), 256, 0, stream>>>(Q, K, V, CTX);
  proj_kernel<<<dim3(NC / 64, (NB * NSEQ) / 128), 256, 0, stream>>>(
      CTX, w_proj, b_proj, out);
}